// GCN_60919816126674
// MI455X (gfx1250) — compile-verified
//
#include <hip/hip_runtime.h>
#include <hip/hip_bf16.h>
#include <math.h>

#define SIZE  256
#define RR    896
#define LMAX  510
#define WMAXX 509
#define HIDC  32
#define ENCC  10
#define GCNH  64
#define NCLS  10
#define BB    8
#define EPSN  7168
#define NNODES (BB*RR)        // 7168
#define NPIX  (3*SIZE*SIZE)   // 196608
#define NE    (BB*EPSN)       // 57344

typedef float v2f __attribute__((ext_vector_type(2)));
typedef float v8f __attribute__((ext_vector_type(8)));

__device__ __forceinline__ float sigmoidf_(float x) { return 1.0f / (1.0f + __expf(-x)); }

// ---------------------------------------------------------------------------
// Stage 1: per-ring conv + decoder einsum as WMMA f32 16x16x4 GEMM
// M = batch (8, padded to 16), N = ENC (10, padded to 16), K = 32*512 (w padded)
// One workgroup (8 waves) per ring; wave wv owns channels c = wv*4 .. wv*4+3.
// Lanes with M>=8 / N>=10 are CLAMPED (not masked): their A/B values are
// duplicates of valid rows, which only pollutes D elements we never read.
// ---------------------------------------------------------------------------
__global__ __launch_bounds__(256) void ring_conv_emb_kernel(
    const float* __restrict__ x, const int* __restrict__ idx,
    const float* __restrict__ Wconv, const float* __restrict__ bconv,
    const float* __restrict__ Wdec, const float* __restrict__ bdec,
    float* __restrict__ nodes)
{
    __shared__ float s_vals[BB][2][516];   // gathered ring pixels, zero padded
    __shared__ float s_red[8 * 256];       // per-wave partial C tiles

    const int r   = blockIdx.x;
    const int tid = threadIdx.x;

    // gather: vals[b][hh][j] = xp[b, idx[r,hh,j]]  (idx==PAD or j>=510 -> 0)
    for (int t = tid; t < BB * 2 * 512; t += 256) {
        const int j  = t & 511;
        const int hh = (t >> 9) & 1;
        const int b  = t >> 10;
        float v = 0.0f;
        if (j < LMAX) {
            const int ix = idx[(r * 2 + hh) * LMAX + j];
            if (ix < NPIX) v = x[b * NPIX + ix];
        }
        s_vals[b][hh][j] = v;
    }
    if (tid < BB * 2 * 4) {   // zero the pad lane j = 512..515
        const int j  = 512 + (tid & 3);
        const int hh = (tid >> 2) & 1;
        const int b  = tid >> 3;
        s_vals[b][hh][j] = 0.0f;
    }
    __syncthreads();

    const int wave = tid >> 5;
    const int lane = tid & 31;
    const int hi   = lane >> 4;                    // K-half per ISA 16x4 f32 layout
    const int mn   = lane & 15;                    // M (A row) and N (B col)
    const int mnA  = (mn < BB)   ? mn : (BB - 1);  // clamp: dup rows are harmless
    const int mnB  = (mn < ENCC) ? mn : (ENCC - 1);

    const float* __restrict__ v0row = &s_vals[mnA][0][0];
    const float* __restrict__ v1row = &s_vals[mnA][1][0];

    v8f acc = {0.f, 0.f, 0.f, 0.f, 0.f, 0.f, 0.f, 0.f};

    for (int cc = 0; cc < 4; ++cc) {
        const int c = wave * 4 + cc;
        const float* wc = Wconv + ((size_t)r * HIDC + c) * 4;
        const float w00 = wc[0], w01 = wc[1], w10 = wc[2], w11 = wc[3];
        const float bc  = bconv[r * HIDC + c];
        const float* __restrict__ wd =
            Wdec + (((size_t)r * ENCC + mnB) * HIDC + c) * (size_t)WMAXX;

        int   w0 = 2 * hi;
        float b0 = wd[w0];
        float b1 = wd[w0 + 1];

        // 127 straight-line iterations, fully in-bounds, B loads pipelined 1 ahead
        for (int w = 0; w < 508; w += 4) {
            const int wn  = w0 + 4;
            const int wnc = (wn > 507) ? 507 : wn;   // clamp; clamped values unused
            const float nb0 = wd[wnc];
            const float nb1 = wd[wnc + 1];

            const float p0 = v0row[w0], p1 = v0row[w0 + 1], p2 = v0row[w0 + 2];
            const float q0 = v1row[w0], q1 = v1row[w0 + 1], q2 = v1row[w0 + 2];
            const float a0 = fmaxf(fmaf(w00, p0, fmaf(w01, p1, fmaf(w10, q0, fmaf(w11, q1, bc)))), 0.f);
            const float a1 = fmaxf(fmaf(w00, p1, fmaf(w01, p2, fmaf(w10, q1, fmaf(w11, q2, bc)))), 0.f);

            v2f av; av.x = a0; av.y = a1;
            v2f bv; bv.x = b0; bv.y = b1;
            acc = __builtin_amdgcn_wmma_f32_16x16x4_f32(false, av, false, bv,
                                                        (short)0, acc, false, false);
            b0 = nb0; b1 = nb1; w0 = wn;
        }

        // masked tail (w0 = 508 + 2*hi), branchless selects, clamped addresses
        {
            const float p0 = v0row[w0], p1 = v0row[w0 + 1], p2 = v0row[w0 + 2];
            const float q0 = v1row[w0], q1 = v1row[w0 + 1], q2 = v1row[w0 + 2];
            float a0 = fmaxf(fmaf(w00, p0, fmaf(w01, p1, fmaf(w10, q0, fmaf(w11, q1, bc)))), 0.f);
            float a1 = fmaxf(fmaf(w00, p1, fmaf(w01, p2, fmaf(w10, q1, fmaf(w11, q2, bc)))), 0.f);
            a0 = (w0 < WMAXX)     ? a0 : 0.f;
            a1 = (w0 + 1 < WMAXX) ? a1 : 0.f;
            const float t0  = wd[(w0 < WMAXX)     ? w0       : (WMAXX - 1)];
            const float t1  = wd[(w0 + 1 < WMAXX) ? (w0 + 1) : (WMAXX - 1)];
            const float tb0 = (w0 < WMAXX)     ? t0 : 0.f;
            const float tb1 = (w0 + 1 < WMAXX) ? t1 : 0.f;

            v2f av; av.x = a0;  av.y = a1;
            v2f bv; bv.x = tb0; bv.y = tb1;
            acc = __builtin_amdgcn_wmma_f32_16x16x4_f32(false, av, false, bv,
                                                        (short)0, acc, false, false);
        }
    }

    // cross-wave reduction of partial 16x16 tiles
#pragma unroll
    for (int v = 0; v < 8; ++v) s_red[wave * 256 + v * 32 + lane] = acc[v];
    __syncthreads();

    if (tid < BB * ENCC) {
        const int b = tid / ENCC, o = tid % ENCC;
        float s = 0.f;
#pragma unroll
        for (int wv = 0; wv < 8; ++wv) s += s_red[wv * 256 + b * 32 + o]; // M=b -> VGPR b, lane o
        nodes[(size_t)(b * RR + r) * ENCC + o] = s + bdec[r * ENCC + o];
    }
}

// ---------------------------------------------------------------------------
// Stage 2: GCN helpers
// ---------------------------------------------------------------------------
__global__ void deg_init_kernel(float* __restrict__ deg) {
    const int i = blockIdx.x * 256 + threadIdx.x;
    if (i < NNODES) deg[i] = 1.0f;   // self-loop weight
}

__global__ void deg_acc_kernel(const int* __restrict__ ei, const float* __restrict__ ew,
                               float* __restrict__ deg) {
    const int e = blockIdx.x * 256 + threadIdx.x;
    if (e < NE) atomicAdd(&deg[ei[NE + e]], sigmoidf_(ew[e % EPSN]));
}

__global__ void dinv_kernel(float* __restrict__ deg) {
    const int i = blockIdx.x * 256 + threadIdx.x;
    if (i < NNODES) {
        const float d = deg[i];
        deg[i] = (d > 0.f) ? rsqrtf(fmaxf(d, 1e-12f)) : 0.f;
    }
}

__global__ void norm_kernel(const int* __restrict__ ei, const float* __restrict__ ew,
                            const float* __restrict__ dinv, float* __restrict__ nrm) {
    const int e = blockIdx.x * 256 + threadIdx.x;
    if (e < NE) nrm[e] = dinv[ei[e]] * sigmoidf_(ew[e % EPSN]) * dinv[ei[NE + e]];
}

// xw1 = nodes(7168x10) @ Wg1(10x64)
__global__ void xw1_kernel(const float* __restrict__ nodes, const float* __restrict__ Wg1,
                           float* __restrict__ xw1) {
    const int t = blockIdx.x * 256 + threadIdx.x;
    if (t < NNODES * GCNH) {
        const int i = t >> 6, f = t & 63;
        float s = 0.f;
#pragma unroll
        for (int k = 0; k < ENCC; ++k) s = fmaf(nodes[i * ENCC + k], Wg1[k * GCNH + f], s);
        xw1[t] = s;
    }
}

// out[i,f] = dinv[i]^2 * xw[i,f] + bias[f]   (self-loop term + bias)
__global__ void self_init_kernel(const float* __restrict__ xw, const float* __restrict__ dinv,
                                 const float* __restrict__ bias, float* __restrict__ out) {
    const int t = blockIdx.x * 256 + threadIdx.x;
    if (t < NNODES * GCNH) {
        const int i = t >> 6, f = t & 63;
        const float di = dinv[i];
        out[t] = fmaf(di * di, xw[t], bias[f]);
    }
}

// out[dst,f] += norm[e] * xw[src,f]
__global__ void scatter_kernel(const int* __restrict__ ei, const float* __restrict__ nrm,
                               const float* __restrict__ xw, float* __restrict__ out) {
    const int t = blockIdx.x * 256 + threadIdx.x;
    if (t < NE * GCNH) {
        const int e = t >> 6, f = t & 63;
        atomicAdd(&out[ei[NE + e] * GCNH + f], nrm[e] * xw[ei[e] * GCNH + f]);
    }
}

// xw2 = relu(out1)(7168x64) @ Wg2(64x64) via WMMA f32 16x16x4; one wave per N-tile
__global__ __launch_bounds__(128) void gemm_relu_wmma_kernel(
    const float* __restrict__ A, const float* __restrict__ W, float* __restrict__ out)
{
    const int tid  = threadIdx.x;
    const int wave = tid >> 5, lane = tid & 31;
    const int hi   = lane >> 4, mn = lane & 15;
    const int m0   = blockIdx.x * 16;
    const int n0   = wave * 16;

    v8f acc = {0.f, 0.f, 0.f, 0.f, 0.f, 0.f, 0.f, 0.f};
#pragma unroll
    for (int kb = 0; kb < GCNH; kb += 4) {
        const int k0 = kb + 2 * hi;
        v2f av, bv;
        av.x = fmaxf(A[(m0 + mn) * GCNH + k0], 0.f);
        av.y = fmaxf(A[(m0 + mn) * GCNH + k0 + 1], 0.f);
        bv.x = W[k0 * GCNH + n0 + mn];
        bv.y = W[(k0 + 1) * GCNH + n0 + mn];
        acc = __builtin_amdgcn_wmma_f32_16x16x4_f32(false, av, false, bv,
                                                    (short)0, acc, false, false);
    }
#pragma unroll
    for (int v = 0; v < 8; ++v)
        out[(m0 + v + 8 * hi) * GCNH + n0 + mn] = acc[v];
}

__global__ void pool_kernel(const float* __restrict__ out2, float* __restrict__ pooled) {
    const int b = blockIdx.x, f = threadIdx.x;
    float s = 0.f;
    for (int r = 0; r < RR; ++r) s += out2[(size_t)(b * RR + r) * GCNH + f];
    pooled[b * GCNH + f] = s * (1.0f / RR);
}

__global__ void head_kernel(const float* __restrict__ pooled, const float* __restrict__ Wl,
                            const float* __restrict__ bl, float* __restrict__ out) {
    __shared__ float s_log[BB][NCLS];
    const int t = threadIdx.x;
    if (t < BB * NCLS) {
        const int b = t / NCLS, o = t % NCLS;
        float s = bl[o];
#pragma unroll
        for (int f = 0; f < GCNH; ++f) s = fmaf(pooled[b * GCNH + f], Wl[f * NCLS + o], s);
        s_log[b][o] = s;
    }
    __syncthreads();
    if (t < BB * NCLS) {
        const int b = t / NCLS, o = t % NCLS;
        float mx = -1e30f;
#pragma unroll
        for (int j = 0; j < NCLS; ++j) mx = fmaxf(mx, s_log[b][j]);
        float den = 0.f;
#pragma unroll
        for (int j = 0; j < NCLS; ++j) den += __expf(s_log[b][j] - mx);
        out[t] = __expf(s_log[b][o] - mx) / den;
    }
}

// ---------------------------------------------------------------------------
extern "C" void kernel_launch(void* const* d_in, const int* in_sizes, int n_in,
                              void* d_out, int out_size, void* d_ws, size_t ws_size,
                              hipStream_t stream) {
    const float* x      = (const float*)d_in[0];
    const int*   eidx   = (const int*)  d_in[1];
    // d_in[2] = batch_vec (derived analytically, unused)
    const int*   idx    = (const int*)  d_in[3];
    const float* Wconv  = (const float*)d_in[4];
    const float* bconv  = (const float*)d_in[5];
    const float* Wdec   = (const float*)d_in[6];
    const float* bdec   = (const float*)d_in[7];
    const float* edge_w = (const float*)d_in[8];
    const float* Wg1    = (const float*)d_in[9];
    const float* bg1    = (const float*)d_in[10];
    const float* Wg2    = (const float*)d_in[11];
    const float* bg2    = (const float*)d_in[12];
    const float* Wl     = (const float*)d_in[13];
    const float* bl     = (const float*)d_in[14];

    // workspace layout (floats)
    float* ws     = (float*)d_ws;
    float* nodes  = ws;                               // 7168*10
    float* dinv   = nodes  + (size_t)NNODES * ENCC;   // 7168 (deg -> dinv in place)
    float* nrm    = dinv   + NNODES;                  // 57344
    float* xw1    = nrm    + NE;                      // 7168*64
    float* out1   = xw1    + (size_t)NNODES * GCNH;
    float* xw2    = out1   + (size_t)NNODES * GCNH;
    float* out2   = xw2    + (size_t)NNODES * GCNH;
    float* pooled = out2   + (size_t)NNODES * GCNH;   // 8*64

    // Stage 1: rings -> node embeddings (WMMA f32)
    ring_conv_emb_kernel<<<RR, 256, 0, stream>>>(x, idx, Wconv, bconv, Wdec, bdec, nodes);

    // Stage 2: GCN
    deg_init_kernel<<<(NNODES + 255) / 256, 256, 0, stream>>>(dinv);
    deg_acc_kernel<<<(NE + 255) / 256, 256, 0, stream>>>(eidx, edge_w, dinv);
    dinv_kernel<<<(NNODES + 255) / 256, 256, 0, stream>>>(dinv);
    norm_kernel<<<(NE + 255) / 256, 256, 0, stream>>>(eidx, edge_w, dinv, nrm);

    xw1_kernel<<<(NNODES * GCNH + 255) / 256, 256, 0, stream>>>(nodes, Wg1, xw1);
    self_init_kernel<<<(NNODES * GCNH + 255) / 256, 256, 0, stream>>>(xw1, dinv, bg1, out1);
    scatter_kernel<<<(NE * GCNH + 255) / 256, 256, 0, stream>>>(eidx, nrm, xw1, out1);

    gemm_relu_wmma_kernel<<<NNODES / 16, 128, 0, stream>>>(out1, Wg2, xw2);
    self_init_kernel<<<(NNODES * GCNH + 255) / 256, 256, 0, stream>>>(xw2, dinv, bg2, out2);
    scatter_kernel<<<(NE * GCNH + 255) / 256, 256, 0, stream>>>(eidx, nrm, xw2, out2);

    pool_kernel<<<BB, GCNH, 0, stream>>>(out2, pooled);
    head_kernel<<<1, 128, 0, stream>>>(pooled, Wl, bl, (float*)d_out);
}